// BiMamba_79250736546334
// MI455X (gfx1250) — compile-verified
//
#include <hip/hip_runtime.h>
#include <cstddef>
#include <cstdint>

// ---------------------------------------------------------------------------
// BiMamba for MI455X (gfx1250): f32 WMMA GEMMs (32x64 wave tiles, fully
// static strides -> immediate-offset loads), async-to-LDS staged selective
// scan, float4 streaming elementwise kernels.
// Shapes: B=2, L=2048, D_MODEL=1024, D_INNER=2048, D_STATE=16, DT_RANK=64.
// ---------------------------------------------------------------------------

typedef __attribute__((ext_vector_type(2))) float v2f;
typedef __attribute__((ext_vector_type(8))) float v8f;

#define BSZ      2
#define SEQ      2048
#define DMODEL   1024
#define DINNER   2048
#define DSTATE   16
#define DTRANK   64
#define XDBL_W   96      // DTRANK + 2*DSTATE
#define ROWS     (BSZ*SEQ)   // 4096 GEMM rows over (b,l)

#define WMMA_F32(a, b, c) \
  __builtin_amdgcn_wmma_f32_16x16x4_f32(false, (a), false, (b), (short)0, (c), false, false)

// Async copy of 4 bytes global -> LDS (CDNA5 GLOBAL_LOAD_ASYNC_TO_LDS_B32,
// tracked with ASYNCcnt). vdst = LDS byte address, vaddr = 64-bit global.
__device__ __forceinline__ void async_copy_f32(uint32_t lds_addr, const float* gsrc) {
  asm volatile("global_load_async_to_lds_b32 %0, %1, off"
               :: "v"(lds_addr), "v"(gsrc) : "memory");
}

// -------------------- GEMM (WMMA f32), 32x64 per wave ----------------------
// C[M x N] = A[M x K] * B^T, Bw is (N x K) row-major. Each wave computes a
// 32x64 tile (2x4 grid of 16x16 WMMA tiles): 6 float2 loads per 8 WMMAs.
// All strides compile-time -> loads use immediate offsets from 2 bases.
template <int LDA, int LDB, int LDC, int K, int NT64>
__global__ __launch_bounds__(256) void gemm_wmma_f32_32x64(
    const float* __restrict__ A, const float* __restrict__ Bw,
    float* __restrict__ C)
{
  const int wave = threadIdx.x >> 5;
  const int lane = threadIdx.x & 31;
  const int tile = blockIdx.x * 8 + wave;
  const int tm = tile / NT64;     // 32-row block
  const int tn = tile % NT64;     // 64-col block
  const int hi = lane >> 4;
  const int lo = lane & 15;

  const float* ap = A  + (size_t)(tm * 32 + lo) * LDA + hi * 2;
  const float* bp = Bw + (size_t)(tn * 64 + lo) * LDB + hi * 2;

  v8f a00 = {}, a01 = {}, a02 = {}, a03 = {};
  v8f a10 = {}, a11 = {}, a12 = {}, a13 = {};
#pragma unroll 2
  for (int k = 0; k < K; k += 4) {
    v2f A0 = *(const v2f*)(ap + k);
    v2f A1 = *(const v2f*)(ap + k + 16 * LDA);
    v2f B0 = *(const v2f*)(bp + k);
    v2f B1 = *(const v2f*)(bp + k + 16 * LDB);
    v2f B2 = *(const v2f*)(bp + k + 32 * LDB);
    v2f B3 = *(const v2f*)(bp + k + 48 * LDB);
    a00 = WMMA_F32(A0, B0, a00);
    a01 = WMMA_F32(A0, B1, a01);
    a02 = WMMA_F32(A0, B2, a02);
    a03 = WMMA_F32(A0, B3, a03);
    a10 = WMMA_F32(A1, B0, a10);
    a11 = WMMA_F32(A1, B1, a11);
    a12 = WMMA_F32(A1, B2, a12);
    a13 = WMMA_F32(A1, B3, a13);
  }
  // C frag: VGPR r -> row m = r + 8*hi, col = lo (coalesced stores)
  float* cp = C + (size_t)(tm * 32 + hi * 8) * LDC + tn * 64 + lo;
#pragma unroll
  for (int r = 0; r < 8; ++r) {
    const int o = r * LDC;
    cp[o]                 = a00[r];
    cp[o + 16]            = a01[r];
    cp[o + 32]            = a02[r];
    cp[o + 48]            = a03[r];
    cp[o + 16 * LDC]      = a10[r];
    cp[o + 16 * LDC + 16] = a11[r];
    cp[o + 16 * LDC + 32] = a12[r];
    cp[o + 16 * LDC + 48] = a13[r];
  }
}

// -------------------- GEMM (WMMA f32), 16x16 per wave ----------------------
// Used for the small-N GEMM (N=96).
template <int LDA, int LDB, int LDC, int K, int NT>
__global__ __launch_bounds__(256) void gemm_wmma_f32(
    const float* __restrict__ A, const float* __restrict__ Bw,
    float* __restrict__ C)
{
  const int wave = threadIdx.x >> 5;
  const int lane = threadIdx.x & 31;
  const int tile = blockIdx.x * 8 + wave;
  const int tm = tile / NT;
  const int tn = tile % NT;
  const int hi = lane >> 4;
  const int lo = lane & 15;

  const float* ap = A  + (size_t)(tm * 16 + lo) * LDA + hi * 2;
  const float* bp = Bw + (size_t)(tn * 16 + lo) * LDB + hi * 2;

  v8f acc = {};
  for (int k = 0; k < K; k += 16) {
#pragma unroll
    for (int kk = 0; kk < 16; kk += 4) {
      v2f a = *(const v2f*)(ap + k + kk);
      v2f b = *(const v2f*)(bp + k + kk);
      acc = WMMA_F32(a, b, acc);
    }
  }
  float* cp = C + (size_t)(tm * 16 + hi * 8) * LDC + tn * 16 + lo;
#pragma unroll
  for (int r = 0; r < 8; ++r) cp[r * LDC] = acc[r];
}

// -------------------- GEMM + (2*bias, softplus) epilogue -------------------
// delta = softplus(dt_lr @ dtp_w^T + 2*bias)   (bias added twice, per ref)
template <int LDA, int LDB, int LDC, int K, int NT>
__global__ __launch_bounds__(256) void gemm_wmma_softplus(
    const float* __restrict__ A, const float* __restrict__ Bw,
    float* __restrict__ C, const float* __restrict__ bias)
{
  const int wave = threadIdx.x >> 5;
  const int lane = threadIdx.x & 31;
  const int tile = blockIdx.x * 8 + wave;
  const int tm = tile / NT;
  const int tn = tile % NT;
  const int hi = lane >> 4;
  const int lo = lane & 15;

  const float* ap = A  + (size_t)(tm * 16 + lo) * LDA + hi * 2;
  const float* bp = Bw + (size_t)(tn * 16 + lo) * LDB + hi * 2;

  v8f acc = {};
  for (int k = 0; k < K; k += 16) {
#pragma unroll
    for (int kk = 0; kk < 16; kk += 4) {
      v2f a = *(const v2f*)(ap + k + kk);
      v2f b = *(const v2f*)(bp + k + kk);
      acc = WMMA_F32(a, b, acc);
    }
  }
  const float b2 = 2.0f * bias[tn * 16 + lo];
  float* cp = C + (size_t)(tm * 16 + hi * 8) * LDC + tn * 16 + lo;
#pragma unroll
  for (int r = 0; r < 8; ++r) {
    float x = acc[r] + b2;
    cp[r * LDC] = (x > 20.0f) ? x : log1pf(__expf(x));
  }
}

// -------------------- depthwise causal conv (k=4) + SiLU -------------------
// Each thread produces 4 consecutive d-channels (b128 traffic).
// dir=0: u_f[b,l,d] = silu(bias + sum_j w[d,j] * xh[b, l-3+j, d])
// dir=1: sequence reversed on input (u_b indexed in reversed time t).
__global__ __launch_bounds__(256) void conv_silu_kernel(
    const float* __restrict__ xz,
    const float* __restrict__ w_f, const float* __restrict__ b_f,
    const float* __restrict__ w_b, const float* __restrict__ b_b,
    float* __restrict__ uf, float* __restrict__ ub)
{
  const int dir = blockIdx.y;
  const size_t i4 = ((size_t)blockIdx.x * 256 + threadIdx.x) * 4;  // (b,l,d)
  const int d = (int)(i4 & (DINNER - 1));
  const int l = (int)((i4 >> 11) & (SEQ - 1));
  const int b = (int)(i4 >> 22);

  const float* w = dir ? w_b : w_f;
  float W[4][4];   // W[c][j] = weight of channel d+c, tap j
  *(float4*)W[0] = *(const float4*)(w + (size_t)(d + 0) * 4);
  *(float4*)W[1] = *(const float4*)(w + (size_t)(d + 1) * 4);
  *(float4*)W[2] = *(const float4*)(w + (size_t)(d + 2) * 4);
  *(float4*)W[3] = *(const float4*)(w + (size_t)(d + 3) * 4);

  float acc[4];
  *(float4*)acc = *(const float4*)((dir ? b_b : b_f) + d);

#pragma unroll
  for (int j = 0; j < 4; ++j) {
    int s = l - 3 + j;
    if (s >= 0) {
      int ls = dir ? (SEQ - 1 - s) : s;
      float xv[4];
      *(float4*)xv = *(const float4*)(xz + ((size_t)b * SEQ + ls) * (2 * DINNER) + d);
#pragma unroll
      for (int c = 0; c < 4; ++c) acc[c] += W[c][j] * xv[c];
    }
  }
  float out[4];
#pragma unroll
  for (int c = 0; c < 4; ++c) out[c] = acc[c] / (1.0f + __expf(-acc[c]));  // SiLU
  *(float4*)((dir ? ub : uf) + i4) = *(float4*)out;
}

// ----------------------------- selective scan ------------------------------
// One lane per (b,d): 16 SSM states in registers, serial over L. A whole
// block shares one batch index b, so the per-step B/C vectors (32 floats)
// are block-uniform: wave 0 stages step l+1 into LDS with
// global_load_async_to_lds_b32 (triple-buffered, ASYNCcnt-tracked) while all
// 8 waves compute step l from LDS. u/delta loads stay coalesced per-lane.
__global__ __launch_bounds__(256) void scan_kernel(
    const float* __restrict__ u, const float* __restrict__ delta,
    const float* __restrict__ xdbl, const float* __restrict__ A_log,
    const float* __restrict__ Dp, float* __restrict__ y)
{
  __shared__ __align__(16) float sBC[3][32];

  const int tid = blockIdx.x * 256 + threadIdx.x;  // 0 .. BSZ*DINNER-1
  const int b = tid >> 11;                          // block-uniform
  const int d = tid & (DINNER - 1);

  float Av[DSTATE];
#pragma unroll
  for (int n = 0; n < DSTATE; ++n) Av[n] = -__expf(A_log[d * DSTATE + n]);
  const float Dd = Dp[d];

  float h[DSTATE];
#pragma unroll
  for (int n = 0; n < DSTATE; ++n) h[n] = 0.0f;

  size_t idx = (size_t)b * SEQ * DINNER + d;
  const float* xd0 = xdbl + (size_t)b * SEQ * XDBL_W + DTRANK;

  // Prime the pipeline: stage B/C for l=0 into buffer 0.
  if (threadIdx.x < 32) {
    async_copy_f32((uint32_t)(uintptr_t)&sBC[0][threadIdx.x], xd0 + threadIdx.x);
  }

  for (int l = 0; l < SEQ; ++l) {
    if (threadIdx.x < 32) {
      if (l + 1 < SEQ) {
        async_copy_f32((uint32_t)(uintptr_t)&sBC[(l + 1) % 3][threadIdx.x],
                       xd0 + (size_t)(l + 1) * XDBL_W + threadIdx.x);
        // oldest outstanding copy (for step l) must complete; keep l+1 in flight
        asm volatile("s_wait_asynccnt 1" ::: "memory");
      } else {
        asm volatile("s_wait_asynccnt 0" ::: "memory");
      }
    }
    __syncthreads();   // buf[l%3] visible to all waves; also fences reuse

    const float* bc = &sBC[l % 3][0];
    float Bv[DSTATE], Cv[DSTATE];
    *(float4*)&Bv[0]  = *(const float4*)(bc + 0);
    *(float4*)&Bv[4]  = *(const float4*)(bc + 4);
    *(float4*)&Bv[8]  = *(const float4*)(bc + 8);
    *(float4*)&Bv[12] = *(const float4*)(bc + 12);
    *(float4*)&Cv[0]  = *(const float4*)(bc + 16);
    *(float4*)&Cv[4]  = *(const float4*)(bc + 20);
    *(float4*)&Cv[8]  = *(const float4*)(bc + 24);
    *(float4*)&Cv[12] = *(const float4*)(bc + 28);

    const float dl = delta[idx];
    const float uv = u[idx];
    const float du = dl * uv;
    float yv = 0.0f;
#pragma unroll
    for (int n = 0; n < DSTATE; ++n) {
      float dA = __expf(dl * Av[n]);
      h[n] = dA * h[n] + du * Bv[n];
      yv += h[n] * Cv[n];
    }
    y[idx] = yv + Dd * uv;   // in-place over u (u[idx] already consumed)

    idx += DINNER;
    if (l + 1 < SEQ) {
      __builtin_prefetch(delta + idx, 0, 0);
      __builtin_prefetch(u + idx, 0, 0);
    }
  }
}

// --------------------------- gate: (yf + flip(yb)) * silu(z) ---------------
// float4 per thread (b128 traffic).
__global__ __launch_bounds__(256) void gate_kernel(
    float* __restrict__ yf, const float* __restrict__ yb,
    const float* __restrict__ xz)
{
  const size_t i4 = ((size_t)blockIdx.x * 256 + threadIdx.x) * 4;
  const int d = (int)(i4 & (DINNER - 1));
  const int l = (int)((i4 >> 11) & (SEQ - 1));
  const int b = (int)(i4 >> 22);

  float z4[4], f4[4], b4[4], o4[4];
  *(float4*)z4 = *(const float4*)(xz + ((size_t)b * SEQ + l) * (2 * DINNER) + DINNER + d);
  *(float4*)f4 = *(const float4*)(yf + i4);
  *(float4*)b4 = *(const float4*)(yb + ((size_t)b * SEQ + (SEQ - 1 - l)) * DINNER + d);
#pragma unroll
  for (int c = 0; c < 4; ++c) {
    float g = z4[c] / (1.0f + __expf(-z4[c]));
    o4[c] = (f4[c] + b4[c]) * g;
  }
  *(float4*)(yf + i4) = *(float4*)o4;
}

// ---------------------------------------------------------------------------
extern "C" void kernel_launch(void* const* d_in, const int* in_sizes, int n_in,
                              void* d_out, int out_size, void* d_ws, size_t ws_size,
                              hipStream_t stream) {
  const float* x       = (const float*)d_in[0];   // (2,2048,1024)
  const float* W_in    = (const float*)d_in[1];   // (4096,1024)
  const float* conv_w  = (const float*)d_in[2];   // (2048,1,4)
  const float* conv_b  = (const float*)d_in[3];   // (2048)
  const float* conv_wb = (const float*)d_in[4];
  const float* conv_bb = (const float*)d_in[5];
  const float* xp_w    = (const float*)d_in[6];   // (96,2048)
  const float* xp_bw   = (const float*)d_in[7];
  const float* dt_w    = (const float*)d_in[8];   // (2048,64)
  const float* dt_b    = (const float*)d_in[9];   // (2048)
  const float* dt_wb   = (const float*)d_in[10];
  const float* dt_bb   = (const float*)d_in[11];
  const float* A_log   = (const float*)d_in[12];  // (2048,16)
  const float* A_b_log = (const float*)d_in[13];
  const float* Dp      = (const float*)d_in[14];  // (2048)
  const float* Dbp     = (const float*)d_in[15];
  const float* W_out   = (const float*)d_in[16];  // (1024,2048)
  float* out = (float*)d_out;                     // (2,2048,1024)

  // Workspace layout (floats)
  float* ws  = (float*)d_ws;
  float* xz  = ws;                                  // (2,2048,4096) 16.78M
  float* uf  = xz  + (size_t)ROWS * 2 * DINNER;     // (2,2048,2048)  8.39M
  float* ub  = uf  + (size_t)ROWS * DINNER;         //                8.39M
  float* xdf = ub  + (size_t)ROWS * DINNER;         // (2,2048,96)    0.39M
  float* xdb = xdf + (size_t)ROWS * XDBL_W;         //                0.39M
  float* df  = xdb + (size_t)ROWS * XDBL_W;         // (2,2048,2048)  8.39M
  float* db  = df  + (size_t)ROWS * DINNER;         //                8.39M

  // 1) xz = x @ in_proj_w^T   M=4096 N=4096 K=1024 -> (128 x 64) 32x64 tiles
  gemm_wmma_f32_32x64<DMODEL, DMODEL, 2 * DINNER, DMODEL, 64>
      <<<8192 / 8, 256, 0, stream>>>(x, W_in, xz);

  // 2) depthwise causal conv + SiLU (fwd + reversed), both directions
  conv_silu_kernel<<<dim3(ROWS * DINNER / 4 / 256, 2), 256, 0, stream>>>(
      xz, conv_w, conv_b, conv_wb, conv_bb, uf, ub);

  // 3) x_dbl = u @ x_proj_w^T   M=4096 N=96 K=2048 -> 1536 16x16 tiles
  gemm_wmma_f32<DINNER, DINNER, XDBL_W, DINNER, 6>
      <<<1536 / 8, 256, 0, stream>>>(uf, xp_w, xdf);
  gemm_wmma_f32<DINNER, DINNER, XDBL_W, DINNER, 6>
      <<<1536 / 8, 256, 0, stream>>>(ub, xp_bw, xdb);

  // 4) delta = softplus(dt_lr @ dt_proj_w^T + 2*bias)  M=4096 N=2048 K=64
  gemm_wmma_softplus<XDBL_W, DTRANK, DINNER, DTRANK, 128>
      <<<32768 / 8, 256, 0, stream>>>(xdf, dt_w, df, dt_b);
  gemm_wmma_softplus<XDBL_W, DTRANK, DINNER, DTRANK, 128>
      <<<32768 / 8, 256, 0, stream>>>(xdb, dt_wb, db, dt_bb);

  // 5) selective scan (writes y in-place over u)
  scan_kernel<<<BSZ * DINNER / 256, 256, 0, stream>>>(uf, df, xdf, A_log, Dp, uf);
  scan_kernel<<<BSZ * DINNER / 256, 256, 0, stream>>>(ub, db, xdb, A_b_log, Dbp, ub);

  // 6) y = (y_f + flip(y_b)) * silu(z)   (in-place into uf)
  gate_kernel<<<ROWS * DINNER / 4 / 256, 256, 0, stream>>>(uf, ub, xz);

  // 7) out = y @ out_proj_w^T   M=4096 N=1024 K=2048 -> (128 x 16) 32x64 tiles
  gemm_wmma_f32_32x64<DINNER, DINNER, DMODEL, DINNER, 16>
      <<<2048 / 8, 256, 0, stream>>>(uf, W_out, out);
}